// QuantumMambaHybridLayer_74431783239897
// MI455X (gfx1250) — compile-verified
//
#include <hip/hip_runtime.h>
#include <math.h>

#define NQ      10
#define DIM     1024          // 2^NQ
#define BATCHSZ 2048
#define FEAT    256
#define NPROJ   80            // 4*NQ*QLCU_LAYERS
#define TWOPI   6.28318530717958647692f

typedef float v2f __attribute__((ext_vector_type(2)));
typedef float v8f __attribute__((ext_vector_type(8)));

// Compiler-level memory ordering fence. Hardware keeps per-wave DS ops
// in order and wave32 lanes are lockstep, so this is sufficient for
// intra-wave LDS producer->consumer across gates.
__device__ __forceinline__ void wsync() { asm volatile("" ::: "memory"); }

__device__ __forceinline__ int ins0(int p, int q) {
  return ((p >> q) << (q + 1)) | (p & ((1 << q) - 1));
}

// ---------------- single-qubit gates (state in LDS, one wave per state) ----

__device__ __forceinline__ void gate_ry(float2* S, int lane, int q, float half) {
  float c = __cosf(half), s = __sinf(half);
  for (int t = 0; t < 16; ++t) {
    int p  = lane + 32 * t;
    int i0 = ins0(p, q), i1 = i0 | (1 << q);
    float2 a0 = S[i0], a1 = S[i1];
    S[i0] = make_float2(c * a0.x - s * a1.x, c * a0.y - s * a1.y);
    S[i1] = make_float2(s * a0.x + c * a1.x, s * a0.y + c * a1.y);
  }
  wsync();
}

__device__ __forceinline__ void gate_rx(float2* S, int lane, int q, float half) {
  float c = __cosf(half), s = __sinf(half);
  for (int t = 0; t < 16; ++t) {
    int p  = lane + 32 * t;
    int i0 = ins0(p, q), i1 = i0 | (1 << q);
    float2 a0 = S[i0], a1 = S[i1];
    // n0 = c*a0 - i s a1 ; n1 = c*a1 - i s a0
    S[i0] = make_float2(c * a0.x + s * a1.y, c * a0.y - s * a1.x);
    S[i1] = make_float2(c * a1.x + s * a0.y, c * a1.y - s * a0.x);
  }
  wsync();
}

__device__ __forceinline__ void gate_rz(float2* S, int lane, int q, float half) {
  float c = __cosf(half), s0 = __sinf(half);
  for (int t = 0; t < 32; ++t) {
    int i = lane + 32 * t;
    float s = ((i >> q) & 1) ? s0 : -s0;       // phase e^{+-i half}
    float2 a = S[i];
    S[i] = make_float2(a.x * c - a.y * s, a.x * s + a.y * c);
  }
  wsync();
}

__device__ __forceinline__ void gate_crz(float2* S, int lane, int ctrl, int tgt, float half) {
  float c = __cosf(half), s0 = __sinf(half);
  for (int t = 0; t < 32; ++t) {
    int i = lane + 32 * t;
    if ((i >> ctrl) & 1) {
      float s = ((i >> tgt) & 1) ? s0 : -s0;
      float2 a = S[i];
      S[i] = make_float2(a.x * c - a.y * s, a.x * s + a.y * c);
    }
  }
  wsync();
}

// ---------------- two-qubit Ising gates: i couples with j = i ^ mask -------

__device__ __forceinline__ void gate_xx(float2* S, int lane, int qa, int qb, float half) {
  float c = __cosf(half), s = __sinf(half);
  int mask = (1 << qa) | (1 << qb);
  for (int t = 0; t < 16; ++t) {
    int p = lane + 32 * t;
    int i = ins0(p, qa);           // bit qa of i is 0 -> each pair visited once
    int j = i ^ mask;
    float2 ai = S[i], aj = S[j];
    S[i] = make_float2(c * ai.x + s * aj.y, c * ai.y - s * aj.x);  // c*ai - i s aj
    S[j] = make_float2(c * aj.x + s * ai.y, c * aj.y - s * ai.x);
  }
  wsync();
}

__device__ __forceinline__ void gate_yy(float2* S, int lane, int qa, int qb, float half) {
  float c = __cosf(half), s = __sinf(half);
  int mask = (1 << qa) | (1 << qb);
  for (int t = 0; t < 16; ++t) {
    int p = lane + 32 * t;
    int i = ins0(p, qa);           // bit qa == 0
    int j = i ^ mask;
    // sign = -1 if bits(qa,qb) equal; bit qa = 0 here; same sign for both ends
    float ss = (((i >> qb) & 1) ? s : -s);
    float2 ai = S[i], aj = S[j];
    S[i] = make_float2(c * ai.x + ss * aj.y, c * ai.y - ss * aj.x);
    S[j] = make_float2(c * aj.x + ss * ai.y, c * aj.y - ss * ai.x);
  }
  wsync();
}

__device__ __forceinline__ void init_state(float2* S, int lane) {
  for (int t = 0; t < 32; ++t) {
    int i = lane + 32 * t;
    S[i] = make_float2(i == 0 ? 1.f : 0.f, 0.f);
  }
  wsync();
}

// measurement -> fused dot with v[30] (order X0..9, Y0..9, Z0..9)
__device__ __forceinline__ float measure_dot(const float2* S, int lane, const float* v) {
  float dot = 0.f;
  for (int w = 0; w < NQ; ++w) {
    float sx = 0.f, sy = 0.f, sz = 0.f;
    for (int t = 0; t < 16; ++t) {
      int p  = lane + 32 * t;
      int i0 = ins0(p, w), i1 = i0 | (1 << w);
      float2 a0 = S[i0], a1 = S[i1];
      sx += a0.x * a1.x + a0.y * a1.y;            // Re(conj(a0)*a1)
      sy += a0.x * a1.y - a0.y * a1.x;            // Im(conj(a0)*a1)
      sz += (a0.x * a0.x + a0.y * a0.y) - (a1.x * a1.x + a1.y * a1.y);
    }
    for (int off = 16; off > 0; off >>= 1) {
      sx += __shfl_xor(sx, off, 32);
      sy += __shfl_xor(sy, off, 32);
      sz += __shfl_xor(sz, off, 32);
    }
    dot += (2.f * sx) * v[w] + (2.f * sy) * v[NQ + w] + sz * v[2 * NQ + w];
  }
  return dot;
}

__device__ __forceinline__ void measure_store(const float2* S, int lane, float* m) {
  for (int w = 0; w < NQ; ++w) {
    float sx = 0.f, sy = 0.f, sz = 0.f;
    for (int t = 0; t < 16; ++t) {
      int p  = lane + 32 * t;
      int i0 = ins0(p, w), i1 = i0 | (1 << w);
      float2 a0 = S[i0], a1 = S[i1];
      sx += a0.x * a1.x + a0.y * a1.y;
      sy += a0.x * a1.y - a0.y * a1.x;
      sz += (a0.x * a0.x + a0.y * a0.y) - (a1.x * a1.x + a1.y * a1.y);
    }
    for (int off = 16; off > 0; off >>= 1) {
      sx += __shfl_xor(sx, off, 32);
      sy += __shfl_xor(sy, off, 32);
      sz += __shfl_xor(sz, off, 32);
    }
    if (lane == 0) { m[w] = 2.f * sx; m[NQ + w] = 2.f * sy; m[2 * NQ + w] = sz; }
  }
  wsync();
}

// ===== Stage 0: batch-independent branches + output folding (one wave) =====
__global__ void stage0_kernel(const float* __restrict__ gate_base,
                              const float* __restrict__ skip_base,
                              const float* __restrict__ W1, const float* __restrict__ b1,
                              const float* __restrict__ W2, const float* __restrict__ b2,
                              const float* __restrict__ W3, const float* __restrict__ b3,
                              const float* __restrict__ w1, const float* __restrict__ w2,
                              const float* __restrict__ w3,
                              const float* __restrict__ W_out, const float* __restrict__ b_out,
                              float* __restrict__ ws) {
  __shared__ float2 S[DIM];
  __shared__ float meas[60];
  int lane = threadIdx.x;

  // gating circuit (2 layers of rx, crz-chain, ry, wrap-crz)
  init_state(S, lane);
  int idx = 0;
  for (int layer = 0; layer < 2; ++layer) {
    for (int i = 0; i < NQ; ++i)     gate_rx(S, lane, i, 0.5f * gate_base[idx++]);
    for (int i = 0; i < NQ - 1; ++i) gate_crz(S, lane, i, i + 1, 0.5f * gate_base[idx++]);
    for (int i = 0; i < NQ; ++i)     gate_ry(S, lane, i, 0.5f * gate_base[idx++]);
    gate_crz(S, lane, NQ - 1, 0, 0.5f * gate_base[idx++]);
  }
  measure_store(S, lane, meas);          // m2

  // skip circuit
  init_state(S, lane);
  for (int i = 0; i < NQ; ++i) gate_rx(S, lane, i, 0.5f * skip_base[i]);
  for (int i = 0; i < NQ; ++i) gate_ry(S, lane, i, 0.5f * skip_base[NQ + i]);
  for (int i = 0; i < NQ; ++i) gate_rz(S, lane, i, 0.5f * skip_base[2 * NQ + i]);
  measure_store(S, lane, meas + 30);     // m3

  float w1v = w1[0], w2v = w2[0], w3v = w3[0];
  if (lane < 30) {                        // v[k] = w1 * sum_j Wout[j]*W1[j][k]
    float vk = 0.f;
    for (int j = 0; j < 30; ++j) vk += W_out[j] * W1[j * 30 + lane];
    ws[lane] = w1v * vk;
  }
  float part = 0.f;
  if (lane < 30) {                        // c0 contribution per j = lane
    float y2 = b2[lane], y3 = b3[lane];
    for (int k = 0; k < 30; ++k) {
      y2 += meas[k] * W2[lane * 30 + k];
      y3 += meas[30 + k] * W3[lane * 30 + k];
    }
    part = W_out[lane] * (w2v * y2 + w3v * y3 + w1v * b1[lane]);
  }
  for (int off = 16; off > 0; off >>= 1) part += __shfl_xor(part, off, 32);
  if (lane == 0) ws[30] = part + b_out[0];
}

// ===== Stage 1: projection GEMM on the matrix units (f32 WMMA 16x16x4) ====
// proj[m][n] = sum_k x[m][k] * W_proj[n][k]; one wave per 16x16 tile.
__global__ void proj_wmma_kernel(const float* __restrict__ x,
                                 const float* __restrict__ Wp,
                                 const float* __restrict__ bproj,
                                 const float* __restrict__ qbase,
                                 float* __restrict__ qlcu) {
  int lane  = threadIdx.x;
  int tile  = blockIdx.x;            // 128 M-tiles x 5 N-tiles
  int mT    = tile / 5, nT = tile % 5;
  int lm    = lane & 15, lh = lane >> 4;

  const float* xr = x  + (size_t)(mT * 16 + lm) * FEAT;  // A row (M = lm)
  const float* wr = Wp + (size_t)(nT * 16 + lm) * FEAT;  // B col (N = lm), K along row

  v8f acc = {0.f, 0.f, 0.f, 0.f, 0.f, 0.f, 0.f, 0.f};
  for (int k = 0; k < FEAT; k += 4) {
    // A lane layout: M = lane%16, K = 2*(lane/16) + reg  -> contiguous float2
    v2f a = *(const v2f*)(xr + k + 2 * lh);
    v2f b = *(const v2f*)(wr + k + 2 * lh);
    acc = __builtin_amdgcn_wmma_f32_16x16x4_f32(false, a, false, b,
                                                (short)0, acc, false, false);
  }

  int   n  = nT * 16 + lm;
  float bp = bproj[n], qb = qbase[n];
#pragma unroll
  for (int r = 0; r < 8; ++r) {            // C/D: M = r + 8*lh, N = lane%16
    int   m   = mT * 16 + r + 8 * lh;
    float p   = acc[r] + bp;
    float sig = 1.f / (1.f + __expf(-p));
    qlcu[(size_t)m * NPROJ + n] = sig * TWOPI + qb;
  }
}

// ===== Stage 2: per-batch statevector simulation, one wave per element ====
__global__ void __launch_bounds__(128)
sim_kernel(const float* __restrict__ qlcu_all,
           const float* __restrict__ bpar,
           const float* __restrict__ cpar,
           const float* __restrict__ dtpar,
           const float* __restrict__ vc,     // ws: v[0..29], c0 at [30]
           float* __restrict__ out) {
  __shared__ float2 Sblk[4 * DIM];          // 32 KB: 4 waves x 8 KB state
  int lane = threadIdx.x & 31;
  int wv   = threadIdx.x >> 5;
  int b    = blockIdx.x * 4 + wv;
  float2* S = Sblk + wv * DIM;

  init_state(S, lane);
  const float* q = qlcu_all + (size_t)b * NPROJ;
  int idx = 0;
  for (int layer = 0; layer < 2; ++layer) {
    for (int i = 0; i < NQ; ++i) gate_ry(S, lane, i, 0.5f * q[idx++]);
    for (int i = 0; i < NQ; ++i) gate_xx(S, lane, i, (i + 1) % NQ, 0.5f * q[idx++]);
    for (int i = 0; i < NQ; ++i) gate_ry(S, lane, i, 0.5f * bpar[i]);
    for (int i = 0; i < NQ; ++i) gate_ry(S, lane, i, 0.5f * q[idx++]);
    for (int i = 0; i < NQ; ++i) gate_yy(S, lane, i, (i + 1) % NQ, 0.5f * q[idx++]);
    for (int i = 0; i < NQ; ++i) gate_rz(S, lane, i, 0.5f * cpar[i]);
    for (int i = 0; i < NQ; ++i) gate_rx(S, lane, i, 0.5f * dtpar[i]);
  }
  float dot = measure_dot(S, lane, vc);
  if (lane == 0) out[b] = dot + vc[30];
}

extern "C" void kernel_launch(void* const* d_in, const int* in_sizes, int n_in,
                              void* d_out, int out_size, void* d_ws, size_t ws_size,
                              hipStream_t stream) {
  const float* x         = (const float*)d_in[0];
  const float* W_proj    = (const float*)d_in[1];
  const float* b_proj    = (const float*)d_in[2];
  const float* qlcu_base = (const float*)d_in[3];
  const float* gate_base = (const float*)d_in[4];
  const float* skip_base = (const float*)d_in[5];
  const float* b_params  = (const float*)d_in[6];
  const float* c_params  = (const float*)d_in[7];
  const float* dt_params = (const float*)d_in[8];
  const float* W1 = (const float*)d_in[9];
  const float* b1 = (const float*)d_in[10];
  const float* W2 = (const float*)d_in[11];
  const float* b2 = (const float*)d_in[12];
  const float* W3 = (const float*)d_in[13];
  const float* b3 = (const float*)d_in[14];
  const float* w1 = (const float*)d_in[15];
  const float* w2 = (const float*)d_in[16];
  const float* w3 = (const float*)d_in[17];
  const float* W_out = (const float*)d_in[18];
  const float* b_out = (const float*)d_in[19];

  float* out  = (float*)d_out;
  float* ws   = (float*)d_ws;
  float* qlcu = ws + 32;                       // 2048*80 floats

  stage0_kernel<<<1, 32, 0, stream>>>(gate_base, skip_base, W1, b1, W2, b2,
                                      W3, b3, w1, w2, w3, W_out, b_out, ws);
  proj_wmma_kernel<<<(BATCHSZ / 16) * 5, 32, 0, stream>>>(x, W_proj, b_proj,
                                                          qlcu_base, qlcu);
  sim_kernel<<<BATCHSZ / 4, 128, 0, stream>>>(qlcu, b_params, c_params,
                                              dt_params, ws, out);
}